// SupConLoss_46531675685457
// MI455X (gfx1250) — compile-verified
//
#include <hip/hip_runtime.h>
#include <hip/hip_bf16.h>
#include <math.h>

typedef __attribute__((ext_vector_type(2))) float v2f;
typedef __attribute__((ext_vector_type(8))) float v8f;

#define HW        9216        // 96*96
#define CCH       128         // channels
#define TILES_M   576         // HW/16
#define MT        2           // M-tiles per wave strip
#define NT        4           // N-tiles per wave strip  -> 32x64 output block
#define STRIPS_M  (TILES_M / MT)          // 288
#define STRIPS_N  (TILES_M / NT)          // 144
#define NSTRIPS   (STRIPS_M * STRIPS_N)   // 41472
#define INV_T     25.0f       // 1/0.04

__global__ void supcon_zero_ws(double* ws) {
    if (threadIdx.x < 2) ws[threadIdx.x] = 0.0;
}

__global__ __launch_bounds__(256)
void supcon_tiles(const float* __restrict__ x, const int* __restrict__ label,
                  double* __restrict__ ws)
{
    const float* __restrict__ x0  = x;             // feat1 as [C][HW]
    const float* __restrict__ x1  = x + CCH * HW;  // feat2 as [C][HW]
    const int*   __restrict__ gt1 = label;         // [HW]
    const int*   __restrict__ gt2 = label + HW;    // [HW]

    const int lane   = threadIdx.x & 31;
    const int wave   = threadIdx.x >> 5;
    const int gwave  = blockIdx.x * (blockDim.x >> 5) + wave;
    const int nwaves = gridDim.x * (blockDim.x >> 5);
    const int lo = lane & 15;   // M/N position within tile
    const int hi = lane >> 4;   // selects K pair for A/B operands

    double pos_acc = 0.0, tot_acc = 0.0;

    // Grid-stride over 32x64 strips; loop bound is wave-uniform so EXEC
    // stays all-ones through every WMMA (ISA requirement).
    for (int s = gwave; s < NSTRIPS; s += nwaves) {
        const int baseM = (s / STRIPS_N) * (16 * MT);
        const int baseN = (s % STRIPS_N) * (16 * NT);

        v8f acc[MT][NT];
        #pragma unroll
        for (int u = 0; u < MT; ++u)
            #pragma unroll
            for (int t = 0; t < NT; ++t) acc[u][t] = (v8f){};

        // K loop: V_WMMA_F32_16X16X4_F32, 32 steps of K=4
        // Per step: 4 A loads + 8 B loads feed 8 WMMAs (1.5 loads/WMMA).
        #pragma unroll 2
        for (int k0 = 0; k0 < CCH; k0 += 4) {
            const int ka = k0 + 2 * hi;
            v2f a[MT];                  // A 16x4: lane holds A[m][ka], A[m][ka+1]
            #pragma unroll
            for (int u = 0; u < MT; ++u) {
                const int m = baseM + u * 16 + lo;
                a[u].x = x0[ka * HW + m];
                a[u].y = x0[(ka + 1) * HW + m];
            }
            v2f b[NT];                  // B 4x16: lane holds B[ka][n], B[ka+1][n]
            #pragma unroll
            for (int t = 0; t < NT; ++t) {
                const int n = baseN + t * 16 + lo;
                b[t].x = x1[ka * HW + n];
                b[t].y = x1[(ka + 1) * HW + n];
            }
            #pragma unroll
            for (int u = 0; u < MT; ++u)
                #pragma unroll
                for (int t = 0; t < NT; ++t)
                    // (neg_a, A, neg_b, B, c_mod, C, reuse_a, reuse_b)
                    acc[u][t] = __builtin_amdgcn_wmma_f32_16x16x4_f32(
                        false, a[u], false, b[t], (short)0, acc[u][t], false, false);
        }

        // Epilogue: exp, label mask, per-strip sums (f32 ok for <=2048 terms)
        float ps = 0.0f, ts = 0.0f;
        int g2[NT];
        #pragma unroll
        for (int t = 0; t < NT; ++t) g2[t] = gt2[baseN + t * 16 + lo];

        #pragma unroll
        for (int u = 0; u < MT; ++u) {
            #pragma unroll
            for (int r = 0; r < 8; ++r) {
                // C/D layout: VGPR r holds row r (lanes 0-15) / row r+8 (lanes 16-31)
                const int g1 = gt1[baseM + u * 16 + r + 8 * hi];
                #pragma unroll
                for (int t = 0; t < NT; ++t) {
                    const float e = __expf(acc[u][t][r] * INV_T);
                    ts += e;
                    if (g1 == g2[t]) ps += e;
                }
            }
        }
        pos_acc += (double)ps;
        tot_acc += (double)ts;
    }

    // wave32 reduction
    #pragma unroll
    for (int off = 16; off > 0; off >>= 1) {
        pos_acc += __shfl_down(pos_acc, off, 32);
        tot_acc += __shfl_down(tot_acc, off, 32);
    }

    __shared__ double sp[8], st[8];
    if (lane == 0) { sp[wave] = pos_acc; st[wave] = tot_acc; }
    __syncthreads();
    if (threadIdx.x == 0) {
        double p = 0.0, t = 0.0;
        const int nw = (int)(blockDim.x >> 5);
        for (int w = 0; w < nw; ++w) { p += sp[w]; t += st[w]; }
        atomicAdd(&ws[0], p);   // global_atomic_add_f64
        atomicAdd(&ws[1], t);
    }
}

__global__ void supcon_finalize(const double* __restrict__ ws, float* __restrict__ out) {
    if (threadIdx.x == 0 && blockIdx.x == 0) {
        const double hw2 = (double)HW * (double)HW;  // 84934656
        out[0] = (float)(-log(ws[0] / ws[1]) / hw2);
    }
}

extern "C" void kernel_launch(void* const* d_in, const int* in_sizes, int n_in,
                              void* d_out, int out_size, void* d_ws, size_t ws_size,
                              hipStream_t stream) {
    const float* x     = (const float*)d_in[0];   // [4,128,96,96] f32
    const int*   label = (const int*)d_in[1];     // [4,1,96,96] i32
    double*      ws    = (double*)d_ws;           // ws[0]=pos_sum, ws[1]=total_sum
    float*       out   = (float*)d_out;

    supcon_zero_ws<<<1, 32, 0, stream>>>(ws);
    supcon_tiles<<<2048, 256, 0, stream>>>(x, label, ws);
    supcon_finalize<<<1, 1, 0, stream>>>(ws, out);
}